// myGraphSAGE_moment_57904749084729
// MI455X (gfx1250) — compile-verified
//
#include <hip/hip_runtime.h>

namespace {

constexpr int   kN   = 100000;   // nodes
constexpr int   kE   = 800000;   // edges
constexpr int   kF   = 128;      // features (layer-1 in, layer-2 in)
constexpr int   kH   = 128;      // hidden
constexpr int   kC   = 40;       // classes
constexpr float kEps = 1e-16f;

typedef __attribute__((ext_vector_type(2))) float v2f;
typedef __attribute__((ext_vector_type(8))) float v8f;

// ---------------------------------------------------------------- degree ----
__global__ __launch_bounds__(256) void deg_count_kernel(
    const int* __restrict__ row, float* __restrict__ deg, int E) {
  int e = blockIdx.x * blockDim.x + threadIdx.x;
  if (e < E) atomicAdd(&deg[row[e]], 1.0f);
}

__global__ __launch_bounds__(256) void deg_inv_kernel(
    float* __restrict__ deg, int n) {
  int i = blockIdx.x * blockDim.x + threadIdx.x;
  if (i < n) {
    float d = deg[i];
    deg[i] = (d > 0.0f) ? (1.0f / fmaxf(d, 1.0f)) : 0.0f;
  }
}

// ---------------------------------------------------- scatter aggregations ----
// acc[row[e]] += src[col[e]]      (one float4 chunk per thread, F = 128)
__global__ __launch_bounds__(256) void scatter_sum_kernel(
    const float* __restrict__ src, const int* __restrict__ row,
    const int* __restrict__ col, float* __restrict__ acc, int E) {
  long long idx = (long long)blockIdx.x * blockDim.x + threadIdx.x;
  long long total = (long long)E * (kF >> 2);
  if (idx >= total) return;
  int e  = (int)(idx >> 5);          // kF/4 == 32 chunks per edge
  int f4 = ((int)idx & 31) << 2;
  int s = col[e], d = row[e];
  const float4 v = *(const float4*)(src + (size_t)s * kF + f4);
  float* p = acc + (size_t)d * kF + f4;
  atomicAdd(p + 0, v.x);
  atomicAdd(p + 1, v.y);
  atomicAdd(p + 2, v.z);
  atomicAdd(p + 3, v.w);
}

// acc[row[e]] += (src[col[e]] - mu[col[e]])^2   (per-source-node centering)
__global__ __launch_bounds__(256) void scatter_var_kernel(
    const float* __restrict__ src, const float* __restrict__ mu,
    const int* __restrict__ row, const int* __restrict__ col,
    float* __restrict__ acc, int E) {
  long long idx = (long long)blockIdx.x * blockDim.x + threadIdx.x;
  long long total = (long long)E * (kF >> 2);
  if (idx >= total) return;
  int e  = (int)(idx >> 5);
  int f4 = ((int)idx & 31) << 2;
  int s = col[e], d = row[e];
  const float4 v = *(const float4*)(src + (size_t)s * kF + f4);
  const float4 m = *(const float4*)(mu  + (size_t)s * kF + f4);
  float dx = v.x - m.x, dy = v.y - m.y, dz = v.z - m.z, dw = v.w - m.w;
  float* p = acc + (size_t)d * kF + f4;
  atomicAdd(p + 0, dx * dx);
  atomicAdd(p + 1, dy * dy);
  atomicAdd(p + 2, dz * dz);
  atomicAdd(p + 3, dw * dw);
}

// mu[i][:] *= deg_inv[i]
__global__ __launch_bounds__(256) void scale_rows_kernel(
    float* __restrict__ a, const float* __restrict__ dinv, int n) {
  long long idx = (long long)blockIdx.x * blockDim.x + threadIdx.x;
  long long total = (long long)n * (kF >> 2);
  if (idx >= total) return;
  int i = (int)(idx >> 5);
  float s = dinv[i];
  float4* p = (float4*)a + idx;
  float4 v = *p;
  v.x *= s; v.y *= s; v.z *= s; v.w *= s;
  *p = v;
}

// sig[i][:] = sqrt( (sig*dinv == 0) ? EPS : sig*dinv )
__global__ __launch_bounds__(256) void finalize_sigma_kernel(
    float* __restrict__ sig, const float* __restrict__ dinv, int n) {
  long long idx = (long long)blockIdx.x * blockDim.x + threadIdx.x;
  long long total = (long long)n * (kF >> 2);
  if (idx >= total) return;
  int i = (int)(idx >> 5);
  float s = dinv[i];
  float4* p = (float4*)sig + idx;
  float4 v = *p;
  v.x *= s; v.y *= s; v.z *= s; v.w *= s;
  v.x = sqrtf(v.x == 0.0f ? kEps : v.x);
  v.y = sqrtf(v.y == 0.0f ? kEps : v.y);
  v.z = sqrtf(v.z == 0.0f ? kEps : v.z);
  v.w = sqrtf(v.w == 0.0f ? kEps : v.w);
  *p = v;
}

// --------------------------------------------------------- dual WMMA GEMM ----
// out[m, c] = act( bias[c] + sum_k A0[m,k]*W0[c,k] + sum_k A1[m,k]*W1[c,k] )
// One block = one 16-row M tile; each wave32 owns a 16x16 output tile at
// col0 = waveId*16 and runs V_WMMA_F32_16X16X4_F32 over K in steps of 4.
// f32 16x16x4 fragments: A lane(l<16): (M=l, K=k0..k0+1), lane(l>=16):
// (M=l-16, K=k0+2..k0+3); B mirrors with N in place of M. C/D: VGPR r holds
// (M = r + 8*(lane>=16), N = lane&15).
__global__ void wmma_dual_gemm_kernel(
    const float* __restrict__ A0, const float* __restrict__ W0,
    const float* __restrict__ A1, const float* __restrict__ W1,
    const float* __restrict__ bias, float* __restrict__ out,
    int K, int nCols, int ldOut, int doRelu) {
  const int m0   = blockIdx.x << 4;
  const int wave = threadIdx.x >> 5;
  const int lane = threadIdx.x & 31;
  const int col0 = wave << 4;
  const int half = lane >> 4;       // 0: K+0/1, 1: K+2/3
  const int l16  = lane & 15;
  const int koff = half << 1;

  const int mrow = m0 + l16;
  const int hcol = col0 + l16;
  const int hcl  = (hcol < nCols) ? hcol : 0;   // clamp OOB weight rows (masked at store)

  const float* a0p = A0 + (size_t)mrow * K + koff;
  const float* b0p = W0 + (size_t)hcl  * K + koff;
  const float* a1p = A1 + (size_t)mrow * K + koff;
  const float* b1p = W1 + (size_t)hcl  * K + koff;

  __builtin_prefetch(a1p, 0, 1);   // global_prefetch_b8: warm 2nd operand stream
  __builtin_prefetch(b1p, 0, 1);

  v8f acc = {};
  for (int k = 0; k < K; k += 4) {
    v2f a = *(const v2f*)(a0p + k);
    v2f b = *(const v2f*)(b0p + k);
    acc = __builtin_amdgcn_wmma_f32_16x16x4_f32(false, a, false, b,
                                                (short)0, acc, false, false);
  }
  for (int k = 0; k < K; k += 4) {
    v2f a = *(const v2f*)(a1p + k);
    v2f b = *(const v2f*)(b1p + k);
    acc = __builtin_amdgcn_wmma_f32_16x16x4_f32(false, a, false, b,
                                                (short)0, acc, false, false);
  }

  const float bv = bias[hcl];
  if (hcol < nCols) {
    const int rbase = m0 + (half << 3);
#pragma unroll
    for (int r = 0; r < 8; ++r) {
      float v = acc[r] + bv;
      if (doRelu) v = fmaxf(v, 0.0f);
      out[(size_t)(rbase + r) * ldOut + hcol] = v;
    }
  }
}

// ------------------------------------------------------------ log-softmax ----
__global__ __launch_bounds__(256) void log_softmax_kernel(
    float* __restrict__ out, int n, int C) {
  int i = blockIdx.x * blockDim.x + threadIdx.x;
  if (i >= n) return;
  float* p = out + (size_t)i * C;
  float mx = -3.402823466e38f;
  for (int c = 0; c < C; ++c) mx = fmaxf(mx, p[c]);
  float s = 0.0f;
  for (int c = 0; c < C; ++c) s += expf(p[c] - mx);
  float l = mx + logf(s);
  for (int c = 0; c < C; ++c) p[c] -= l;
}

}  // namespace

// ------------------------------------------------------------------ driver ----
extern "C" void kernel_launch(void* const* d_in, const int* in_sizes, int n_in,
                              void* d_out, int out_size, void* d_ws, size_t ws_size,
                              hipStream_t stream) {
  const float* x    = (const float*)d_in[0];
  const float* W_l1 = (const float*)d_in[1];
  const float* b_l1 = (const float*)d_in[2];
  const float* W_r1 = (const float*)d_in[3];
  const float* W_l2 = (const float*)d_in[4];
  const float* b_l2 = (const float*)d_in[5];
  const float* W_r2 = (const float*)d_in[6];
  const int*   ei   = (const int*)d_in[7];
  const int* row = ei;        // edge_index[0]
  const int* col = ei + kE;   // edge_index[1]

  // workspace layout (floats): deg_inv | mu | sig | h  -> ~154 MB
  float* deg = (float*)d_ws;
  float* mu  = deg + kN;
  float* sig = mu + (size_t)kN * kF;
  float* h1  = sig + (size_t)kN * kF;
  float* outp = (float*)d_out;                 // [kN, kC]

  const size_t featBytes = sizeof(float) * (size_t)kN * kF;
  const long long edgeChunks = (long long)kE * (kF >> 2);   // 25.6M
  const long long nodeChunks = (long long)kN * (kF >> 2);   // 3.2M
  const int eBlocks  = (int)((edgeChunks + 255) / 256);
  const int nBlocks  = (int)((nodeChunks + 255) / 256);

  // ---- degree ----
  hipMemsetAsync(deg, 0, sizeof(float) * kN, stream);
  deg_count_kernel<<<(kE + 255) / 256, 256, 0, stream>>>(row, deg, kE);
  deg_inv_kernel<<<(kN + 255) / 256, 256, 0, stream>>>(deg, kN);

  // ---- layer 1: mu, sigma, dual GEMM + relu ----
  hipMemsetAsync(mu,  0, featBytes, stream);
  hipMemsetAsync(sig, 0, featBytes, stream);
  scatter_sum_kernel<<<eBlocks, 256, 0, stream>>>(x, row, col, mu, kE);
  scale_rows_kernel<<<nBlocks, 256, 0, stream>>>(mu, deg, kN);
  scatter_var_kernel<<<eBlocks, 256, 0, stream>>>(x, mu, row, col, sig, kE);
  finalize_sigma_kernel<<<nBlocks, 256, 0, stream>>>(sig, deg, kN);
  // 8 waves x 16 cols = 128 output cols per 16-row block
  wmma_dual_gemm_kernel<<<kN / 16, 256, 0, stream>>>(
      sig, W_l1, x, W_r1, b_l1, h1, kF, kH, kH, /*relu=*/1);

  // ---- layer 2: mu, sigma on h, dual GEMM (40 cols) ----
  hipMemsetAsync(mu,  0, featBytes, stream);
  hipMemsetAsync(sig, 0, featBytes, stream);
  scatter_sum_kernel<<<eBlocks, 256, 0, stream>>>(h1, row, col, mu, kE);
  scale_rows_kernel<<<nBlocks, 256, 0, stream>>>(mu, deg, kN);
  scatter_var_kernel<<<eBlocks, 256, 0, stream>>>(h1, mu, row, col, sig, kE);
  finalize_sigma_kernel<<<nBlocks, 256, 0, stream>>>(sig, deg, kN);
  // 3 waves cover cols 0..47; cols >= 40 clamped on load, masked on store
  wmma_dual_gemm_kernel<<<kN / 16, 96, 0, stream>>>(
      sig, W_l2, h1, W_r2, b_l2, outp, kH, kC, kC, /*relu=*/0);

  // ---- log-softmax over 40 classes, in place ----
  log_softmax_kernel<<<(kN + 255) / 256, 256, 0, stream>>>(outp, kN, kC);
}